// RNNDecoder_19086834663739
// MI455X (gfx1250) — compile-verified
//
#include <hip/hip_runtime.h>
#include <math.h>

#define H 2048
#define V 50257

typedef float v2f __attribute__((ext_vector_type(2)));
typedef float v4f __attribute__((ext_vector_type(4)));
typedef float v8f __attribute__((ext_vector_type(8)));

// ---------------------------------------------------------------------------
// Fused batch-1 matvec kernel using V_WMMA_F32_16X16X4_F32.
// Each wave computes 16 output rows: y[m] = dot(W[m,:], x) + bias[m].
// B operand broadcasts x into all 16 columns, so every column of D holds y[m];
// lane 0 ends with y[rowbase+0..7] in acc[0..7], lane 16 with y[rowbase+8..15].
// K mapping per WMMA: lanes 0-15 carry K slots {0,1} = memory cols {k,k+1},
// lanes 16-31 carry K slots {2,3} = memory cols {k+4,k+5} (order-invariant dot),
// so a single per-lane b128 load (col offset k + 4*(lane>=16)) feeds 2 WMMAs.
// ---------------------------------------------------------------------------
__global__ __launch_bounds__(256) void gemv3_wmma(
    const float* __restrict__ w_ih, const float* __restrict__ word,
    const float* __restrict__ b_ih, float* __restrict__ gi,
    const float* __restrict__ w_hh, const float* __restrict__ hid,
    const float* __restrict__ b_hh, float* __restrict__ gh,
    const float* __restrict__ lin_w, const float* __restrict__ lin_b,
    float* __restrict__ logits)
{
    __shared__ __align__(16) float xs[H];

    const int NB_G = (3 * H) / 128;   // 48 blocks per gate matvec
    int b = blockIdx.x;
    const float* W; const float* x; const float* bias; float* out; int nrows; int tile;
    if (b < NB_G)            { W = w_ih;  x = word; bias = b_ih;  out = gi;     nrows = 3 * H; tile = b; }
    else if (b < 2 * NB_G)   { W = w_hh;  x = hid;  bias = b_hh;  out = gh;     nrows = 3 * H; tile = b - NB_G; }
    else                     { W = lin_w; x = hid;  bias = lin_b; out = logits; nrows = V;     tile = b - 2 * NB_G; }

    int t = threadIdx.x;

    // Stage the 8KB vector into LDS (256 threads x 8 floats, b128 each way).
    {
        const v4f* xv = (const v4f*)x;
        v4f*       sv = (v4f*)xs;
        sv[t]       = xv[t];
        sv[t + 256] = xv[t + 256];
    }
    __syncthreads();

    int wave = t >> 5;          // 0..7
    int lane = t & 31;
    int hf   = lane >> 4;       // half-wave: 0 or 1
    int m    = lane & 15;       // row within tile for A operand
    int rowbase = tile * 128 + wave * 16;
    if (rowbase >= nrows) return;          // wave-uniform early out (tail tiles)

    int row = rowbase + m;
    if (row >= nrows) row = nrows - 1;     // clamp for loads; stores are masked

    const float* wrow = W  + (size_t)row * H + 4 * hf;
    const float* xcol = xs + 4 * hf;       // lanes 0-15 broadcast, 16-31 broadcast

    v8f acc = {};
    #pragma unroll 4
    for (int k = 0; k < H; k += 8) {
        v4f a  = *(const v4f*)(wrow + k);  // global_load_b128: W cols k+4*hf .. +3
        v4f bx = *(const v4f*)(xcol + k);  // ds_load_b128 (broadcast within half)
        v2f a0 = { a.x,  a.y  }, a1 = { a.z,  a.w  };
        v2f b0 = { bx.x, bx.y }, b1 = { bx.z, bx.w };
        acc = __builtin_amdgcn_wmma_f32_16x16x4_f32(false, a0, false, b0,
                                                    (short)0, acc, false, false);
        acc = __builtin_amdgcn_wmma_f32_16x16x4_f32(false, a1, false, b1,
                                                    (short)0, acc, false, false);
    }

    // D layout: lane 0 (N=0) holds M=0..7 in acc[0..7]; lane 16 holds M=8..15.
    if (lane == 0 || lane == 16) {
        int r0 = rowbase + 8 * hf;
        #pragma unroll
        for (int i = 0; i < 8; ++i) {
            int r = r0 + i;
            if (r < nrows) out[r] = acc[i] + bias[r];
        }
    }
}

// ---------------------------------------------------------------------------
// GRU gate combine: h_new = (1-z)*n + z*h0
// ---------------------------------------------------------------------------
__global__ __launch_bounds__(256) void gru_combine(
    const float* __restrict__ gi, const float* __restrict__ gh,
    const float* __restrict__ hid, float* __restrict__ hnew)
{
    int j = blockIdx.x * blockDim.x + threadIdx.x;
    if (j >= H) return;
    float r = 1.0f / (1.0f + expf(-(gi[j]         + gh[j])));
    float z = 1.0f / (1.0f + expf(-(gi[H + j]     + gh[H + j])));
    float n = tanhf(gi[2 * H + j] + r * gh[2 * H + j]);
    hnew[j] = (1.0f - z) * n + z * hid[j];
}

// ---------------------------------------------------------------------------
// Softmax over V=50257 logits: 2-stage max, exp+partial-sum, finish, scale.
// ---------------------------------------------------------------------------
__global__ __launch_bounds__(256) void partial_max(const float* __restrict__ logits,
                                                   float* __restrict__ pmax)
{
    __shared__ float s[256];
    int t = threadIdx.x;
    float mx = -INFINITY;
    for (int i = blockIdx.x * 256 + t; i < V; i += gridDim.x * 256)
        mx = fmaxf(mx, logits[i]);
    s[t] = mx; __syncthreads();
    for (int o = 128; o > 0; o >>= 1) {
        if (t < o) s[t] = fmaxf(s[t], s[t + o]);
        __syncthreads();
    }
    if (t == 0) pmax[blockIdx.x] = s[0];
}

__global__ __launch_bounds__(128) void finish_max(const float* __restrict__ pmax,
                                                  float* __restrict__ scal)
{
    __shared__ float s[128];
    int t = threadIdx.x;
    s[t] = pmax[t]; __syncthreads();
    for (int o = 64; o > 0; o >>= 1) {
        if (t < o) s[t] = fmaxf(s[t], s[t + o]);
        __syncthreads();
    }
    if (t == 0) scal[0] = s[0];
}

__global__ __launch_bounds__(256) void exp_partial(const float* __restrict__ logits,
                                                   const float* __restrict__ scal,
                                                   float* __restrict__ e_out,
                                                   float* __restrict__ psum)
{
    __shared__ float s[256];
    float gmax = scal[0];
    int t = threadIdx.x;
    float acc = 0.0f;
    for (int i = blockIdx.x * 256 + t; i < V; i += gridDim.x * 256) {
        float e = expf(logits[i] - gmax);
        e_out[i] = e;
        acc += e;
    }
    s[t] = acc; __syncthreads();
    for (int o = 128; o > 0; o >>= 1) {
        if (t < o) s[t] += s[t + o];
        __syncthreads();
    }
    if (t == 0) psum[blockIdx.x] = s[0];
}

__global__ __launch_bounds__(128) void finish_sum(const float* __restrict__ psum,
                                                  float* __restrict__ scal)
{
    __shared__ float s[128];
    int t = threadIdx.x;
    s[t] = psum[t]; __syncthreads();
    for (int o = 64; o > 0; o >>= 1) {
        if (t < o) s[t] += s[t + o];
        __syncthreads();
    }
    if (t == 0) scal[1] = 1.0f / s[0];
}

__global__ __launch_bounds__(256) void scale_probs(float* __restrict__ out,
                                                   const float* __restrict__ scal)
{
    float inv = scal[1];
    for (int i = blockIdx.x * 256 + threadIdx.x; i < V; i += gridDim.x * 256)
        out[i] *= inv;
}

// ---------------------------------------------------------------------------
extern "C" void kernel_launch(void* const* d_in, const int* in_sizes, int n_in,
                              void* d_out, int out_size, void* d_ws, size_t ws_size,
                              hipStream_t stream)
{
    const float* word   = (const float*)d_in[0];
    const float* hidden = (const float*)d_in[1];
    const float* w_ih   = (const float*)d_in[2];
    const float* w_hh   = (const float*)d_in[3];
    const float* b_ih   = (const float*)d_in[4];
    const float* b_hh   = (const float*)d_in[5];
    const float* lin_w  = (const float*)d_in[6];
    const float* lin_b  = (const float*)d_in[7];

    float* out = (float*)d_out;            // [0..V): probs, [V..V+H): h_new
    float* ws  = (float*)d_ws;

    float* gi     = ws;                    // 6144 floats
    float* gh     = ws + 6144;             // 6144 floats
    float* logits = ws + 12288;            // 50257 floats (ends 62545)
    float* pmax   = ws + 63232;            // 128 floats
    float* psum   = ws + 63360;            // 128 floats
    float* scal   = ws + 63488;            // 2 floats

    const int NB_G = (3 * H) / 128;        // 48
    const int NB_L = (V + 127) / 128;      // 393
    const int nb   = 2 * NB_G + NB_L;      // 489 blocks

    gemv3_wmma<<<nb, 256, 0, stream>>>(w_ih, word, b_ih, gi,
                                       w_hh, hidden, b_hh, gh,
                                       lin_w, lin_b, logits);
    gru_combine<<<(H + 255) / 256, 256, 0, stream>>>(gi, gh, hidden, out + V);
    partial_max<<<128, 256, 0, stream>>>(logits, pmax);
    finish_max<<<1, 128, 0, stream>>>(pmax, scal);
    exp_partial<<<128, 256, 0, stream>>>(logits, scal, out, psum);
    finish_sum<<<1, 128, 0, stream>>>(psum, scal);
    scale_probs<<<128, 256, 0, stream>>>(out, scal);
}